// C2F_model_35038343201527
// MI455X (gfx1250) — compile-verified
//
#include <hip/hip_runtime.h>
#include <hip/hip_bf16.h>

// ---------------- problem constants (fixed by the reference) ----------------
#define Tn    1024          // tokens
#define En    768           // encoder hidden
#define Fn    20            // feature emb size
#define Hn    1000          // ffnn size
#define Wn    20            // max span width
#define SPANn 2324          // 3*E + F
#define KTOP  409           // num_top_spans (reference fixes 409; out_size = 409*409)
#define NCAND (Tn * Wn)     // 20480
#define NSORT 32768         // pow2 >= NCAND
#define PAIRLD 416          // 26 tiles of 16

typedef float v2f __attribute__((ext_vector_type(2)));
typedef float v8f __attribute__((ext_vector_type(8)));

// ---------------- wave32 reductions ----------------
__device__ __forceinline__ float waveSum(float v) {
  for (int o = 16; o > 0; o >>= 1) v += __shfl_xor(v, o, 32);
  return v;
}
__device__ __forceinline__ float waveMax(float v) {
  for (int o = 16; o > 0; o >>= 1) v = fmaxf(v, __shfl_xor(v, o, 32));
  return v;
}

__device__ __forceinline__ v8f wmma4(v2f a, v2f b, v8f c) {
  return __builtin_amdgcn_wmma_f32_16x16x4_f32(false, a, false, b,
                                               (short)0, c, false, false);
}

// ---------------- generic f32 WMMA GEMM: C[M,N] = A[M,K] @ B(+bias) ----------
// One wave computes a 32x32 C macro-tile (2x2 of 16x16 V_WMMA_F32_16X16X4_F32
// tiles, 4 accumulators) so each K-step's 2 A-loads + 2 B-loads feed 4 WMMAs.
// Bounds handling: row/col addresses are CLAMPED (always-legal loads). That is
// safe because a wrong A row only corrupts the same C row and a wrong B column
// only corrupts the same C column, both of which the store guard skips.
// A: row-major [M,lda]. transB=0: B row-major [K,ldb]. transB=1: B row-major
// [N,ldb] (C = A @ B^T). K must be a multiple of 4; ld* must be even.
__global__ void gemm_wmma_f32(const float* __restrict__ A,
                              const float* __restrict__ B,
                              const float* __restrict__ bias,
                              float* __restrict__ C,
                              int M, int N, int K,
                              int lda, int ldb, int ldc, int transB) {
  const int wave = threadIdx.x >> 5;
  const int lane = threadIdx.x & 31;
  const int tilesM = (M + 31) >> 5;
  const int tilesN = (N + 31) >> 5;
  const int tile = blockIdx.x * (blockDim.x >> 5) + wave;
  if (tile >= tilesM * tilesN) return;
  const int m0 = (tile / tilesN) << 5;
  const int n0 = (tile % tilesN) << 5;

  const int half = lane >> 4;     // A/B frag: lanes 0-15 K={0,1}, 16-31 K={2,3}
  const int l16  = lane & 15;

  int ar0 = m0 + l16;       if (ar0 > M - 1) ar0 = M - 1;
  int ar1 = m0 + 16 + l16;  if (ar1 > M - 1) ar1 = M - 1;
  int bc0 = n0 + l16;       if (bc0 > N - 1) bc0 = N - 1;
  int bc1 = n0 + 16 + l16;  if (bc1 > N - 1) bc1 = N - 1;

  const float* a0p = A + (size_t)ar0 * lda + (half << 1);
  const float* a1p = A + (size_t)ar1 * lda + (half << 1);

  v8f acc00 = {}, acc01 = {}, acc10 = {}, acc11 = {};

  if (transB) {
    const float* b0p = B + (size_t)bc0 * ldb + (half << 1);
    const float* b1p = B + (size_t)bc1 * ldb + (half << 1);
#pragma unroll 4
    for (int k = 0; k < K; k += 4) {
      const v2f a0 = *(const v2f*)(a0p + k);
      const v2f a1 = *(const v2f*)(a1p + k);
      const v2f b0 = *(const v2f*)(b0p + k);
      const v2f b1 = *(const v2f*)(b1p + k);
      acc00 = wmma4(a0, b0, acc00);
      acc01 = wmma4(a0, b1, acc01);
      acc10 = wmma4(a1, b0, acc10);
      acc11 = wmma4(a1, b1, acc11);
    }
  } else {
    const float* bbase = B + (size_t)(half << 1) * ldb;
#pragma unroll 4
    for (int k = 0; k < K; k += 4) {
      const v2f a0 = *(const v2f*)(a0p + k);
      const v2f a1 = *(const v2f*)(a1p + k);
      const float* bk = bbase + (size_t)k * ldb;
      v2f b0, b1;
      b0.x = bk[bc0];
      b0.y = bk[(size_t)ldb + bc0];
      b1.x = bk[bc1];
      b1.y = bk[(size_t)ldb + bc1];
      acc00 = wmma4(a0, b0, acc00);
      acc01 = wmma4(a0, b1, acc01);
      acc10 = wmma4(a1, b0, acc10);
      acc11 = wmma4(a1, b1, acc11);
    }
  }

  // store (with optional per-column bias); cols/rows beyond M/N are skipped
  const int col0 = n0 + l16;
  const int col1 = n0 + 16 + l16;
  const float bv0 = (bias && col0 < N) ? bias[col0] : 0.0f;
  const float bv1 = (bias && col1 < N) ? bias[col1] : 0.0f;
#pragma unroll
  for (int v = 0; v < 8; ++v) {
    const int r0 = m0 + v + (half << 3);        // C/D: VGPR v -> M=v (+8 upper)
    const int r1 = r0 + 16;
    if (r0 < M && col0 < N) C[(size_t)r0 * ldc + col0] = acc00[v] + bv0;
    if (r0 < M && col1 < N) C[(size_t)r0 * ldc + col1] = acc01[v] + bv1;
    if (r1 < M && col0 < N) C[(size_t)r1 * ldc + col0] = acc10[v] + bv0;
    if (r1 < M && col1 < N) C[(size_t)r1 * ldc + col1] = acc11[v] + bv1;
  }
}

// ---------------- token attention logits: ta[t] = token_emb[t]·attn_w + b ----
__global__ void token_attn_k(const float* __restrict__ te,
                             const float* __restrict__ aw,
                             const float* __restrict__ ab,
                             float* __restrict__ ta) {
  const int wave = threadIdx.x >> 5, lane = threadIdx.x & 31;
  const int t = blockIdx.x * (blockDim.x >> 5) + wave;
  if (t >= Tn) return;
  float acc = 0.0f;
  for (int e = lane; e < En; e += 32) acc += te[(size_t)t * En + e] * aw[e];
  acc = waveSum(acc);
  if (lane == 0) ta[t] = acc + ab[0];
}

// ---------------- widthW[w,j] = emb_span_width[w]·ms_w1[1536+f, j] ----------
__global__ void width_proj_k(const float* __restrict__ esw,
                             const float* __restrict__ msw1,
                             float* __restrict__ widthW) {
  const int w = blockIdx.x;
  for (int j = threadIdx.x; j < Hn; j += blockDim.x) {
    float acc = 0.0f;
#pragma unroll
    for (int f = 0; f < Fn; ++f)
      acc += esw[w * Fn + f] * msw1[(size_t)(2 * En + f) * Hn + j];
    widthW[(size_t)w * Hn + j] = acc;
  }
}

// ---------------- width-prior scores: wscore[w] ----------------
__global__ void wscore_k(const float* __restrict__ eswp,
                         const float* __restrict__ wsw1,
                         const float* __restrict__ wsb1,
                         const float* __restrict__ wsw2,
                         const float* __restrict__ wsb2,
                         float* __restrict__ wscore) {
  const int w = blockIdx.x;
  float part = 0.0f;
  for (int j = threadIdx.x; j < Hn; j += blockDim.x) {
    float h = wsb1[j];
#pragma unroll
    for (int f = 0; f < Fn; ++f) h += eswp[w * Fn + f] * wsw1[(size_t)f * Hn + j];
    part += fmaxf(h, 0.0f) * wsw2[j];
  }
  __shared__ float red[256];
  red[threadIdx.x] = part;
  __syncthreads();
  for (int s = 128; s > 0; s >>= 1) {
    if (threadIdx.x < s) red[threadIdx.x] += red[threadIdx.x + s];
    __syncthreads();
  }
  if (threadIdx.x == 0) wscore[w] = red[0] + wsb2[0];
}

// ---------------- fused per-span mention score (one wave per span) ----------
__global__ void span_score_k(const float* __restrict__ ta,
                             const float* __restrict__ Q,
                             const float* __restrict__ R,
                             const float* __restrict__ P,
                             const float* __restrict__ widthW,
                             const float* __restrict__ wscore,
                             const float* __restrict__ msb1,
                             const float* __restrict__ msw2,
                             const float* __restrict__ msb2,
                             const int*   __restrict__ smap,
                             float* __restrict__ scores) {
  const int wave = threadIdx.x >> 5, lane = threadIdx.x & 31;
  const int n = blockIdx.x * (blockDim.x >> 5) + wave;
  if (n >= NCAND) return;
  const int s = n / Wn, w = n % Wn;
  const int eRaw = s + w;
  const int e = (eRaw < Tn - 1) ? eRaw : Tn - 1;
  const bool valid = (eRaw < Tn) && (smap[s] == smap[e]);
  if (!valid) { if (lane == 0) scores[n] = -1e9f; return; }

  const int L = w + 1;                         // tokens in span (<= 20)
  // softmax over the span's token_attn values (lanes >= L contribute 0 weight)
  float a = (lane < L) ? ta[s + lane] : -INFINITY;
  const float m = waveMax(a);
  const float ex = (lane < L) ? __expf(a - m) : 0.0f;
  const float ssum = waveSum(ex);
  const float wt = ex / ssum;
  float wts[Wn];                               // wts[t] == 0 for t >= L
#pragma unroll
  for (int t = 0; t < Wn; ++t) wts[t] = __shfl(wt, t, 32);

  float acc = 0.0f;
  for (int j = lane; j < Hn; j += 32) {
    float h = msb1[j] + Q[(size_t)s * Hn + j] + R[(size_t)e * Hn + j]
            + widthW[(size_t)w * Hn + j];
#pragma unroll
    for (int t = 0; t < Wn; ++t) {
      const int tt = (s + t < Tn - 1) ? (s + t) : (Tn - 1);  // clamped; wt=0 beyond L
      h += wts[t] * P[(size_t)tt * Hn + j];
    }
    acc += fmaxf(h, 0.0f) * msw2[j];
  }
  acc = waveSum(acc);
  if (lane == 0) scores[n] = acc + msb2[0] + wscore[w];
}

// ---------------- full bitonic argsort in 256 KB LDS (one workgroup) --------
// key = (~orderedFloat(score)) << 32 | idx  -> ascending sort == stable
// descending argsort of scores (matches jnp.argsort(-scores)).
__global__ void sort_k(const float* __restrict__ scores, int* __restrict__ order) {
  extern __shared__ unsigned long long keys[];
  const int tid = threadIdx.x, nth = blockDim.x;
  for (int i = tid; i < NSORT; i += nth) {
    unsigned long long kk;
    if (i < NCAND) {
      unsigned b = __float_as_uint(scores[i]);
      unsigned ord = b ^ ((b & 0x80000000u) ? 0xFFFFFFFFu : 0x80000000u);
      kk = ((unsigned long long)(~ord) << 32) | (unsigned)i;
    } else {
      kk = ~0ull;
    }
    keys[i] = kk;
  }
  __syncthreads();
  for (unsigned size = 2; size <= NSORT; size <<= 1) {
    for (unsigned stride = size >> 1; stride > 0; stride >>= 1) {
      for (unsigned i = tid; i < NSORT; i += nth) {
        const unsigned j = i ^ stride;
        if (j > i) {
          const unsigned long long a = keys[i], b = keys[j];
          const bool up = ((i & size) == 0);
          if (up ? (a > b) : (a < b)) { keys[i] = b; keys[j] = a; }
        }
      }
      __syncthreads();
    }
  }
  for (int i = tid; i < NCAND; i += nth)
    order[i] = (int)(keys[i] & 0xFFFFFFFFull);
}

// ---------------- greedy non-crossing NMS (single wave, sequential) ---------
__global__ void nms_k(const int* __restrict__ order,
                      const int* __restrict__ smap,
                      int* __restrict__ selRaw, int* __restrict__ countOut) {
  __shared__ int s2me[Tn], e2ms[Tn];
  const int lane = threadIdx.x;
  for (int i = lane; i < Tn; i += 32) { s2me[i] = -1; e2ms[i] = -1; }
  __syncthreads();
  int count = 0;
  for (int c = 0; c < NCAND && count < KTOP; ++c) {
    const int cid = order[c];
    const int s = cid / Wn, w = cid % Wn;
    const int eRaw = s + w;
    const int e = (eRaw < Tn - 1) ? eRaw : Tn - 1;
    if (!((eRaw < Tn) && (smap[s] == smap[e]))) continue;   // invalid (uniform)
    const int len = e - s;                                  // 0..19
    bool bad = false;
    if (lane < len) {
      const int t1 = s + 1 + lane;          // tokens in (s, e]
      bad = s2me[t1] > e;
      const int v = e2ms[s + lane];         // tokens in [s, e)
      bad = bad || (v >= 0 && v < s);
    }
    if (__any(bad)) continue;
    if (lane == 0) {
      if (e > s2me[s]) s2me[s] = e;
      const int v = e2ms[e];
      e2ms[e] = (v < 0) ? s : (v < s ? v : s);
      selRaw[count] = cid;
    }
    ++count;
    __syncthreads();   // make lane 0's LDS updates visible before next reads
  }
  if (lane == 0) countOut[0] = count;
}

// ---------------- sort selected spans by (start,end), pad with first --------
__global__ void selsort_k(const int* __restrict__ selRaw,
                          const int* __restrict__ countPtr,
                          const float* __restrict__ scores,
                          int* __restrict__ selFinal,
                          float* __restrict__ topScores) {
  __shared__ int keys[KTOP], vals[KTOP], sorted[KTOP];
  const int count = countPtr[0];
  const int tid = threadIdx.x;
  if (tid < KTOP) {
    const int cid = (tid < count) ? selRaw[tid] : 0;
    const int s = cid / Wn, w = cid % Wn;
    const int eRaw = s + w;
    const int e = (eRaw < Tn - 1) ? eRaw : Tn - 1;
    keys[tid] = (tid < count) ? (s * Tn + e) : (1 << 30);
    vals[tid] = cid;
  }
  __syncthreads();
  if (tid < KTOP) {
    const int k = keys[tid];
    int r = 0;
    for (int j = 0; j < KTOP; ++j) {
      const int kj = keys[j];
      r += (kj < k) || (kj == k && j < tid);   // stable rank
    }
    sorted[r] = vals[tid];
  }
  __syncthreads();
  if (tid < KTOP) {
    const int cid = (tid < count) ? sorted[tid] : sorted[0];
    selFinal[tid] = cid;
    topScores[tid] = scores[cid];
  }
}

// ---------------- rebuild span embeddings for the 409 winners ---------------
__global__ void topemb_k(const int* __restrict__ selFinal,
                         const float* __restrict__ te,
                         const float* __restrict__ ta,
                         const float* __restrict__ esw,
                         float* __restrict__ topEmb) {
  const int row = blockIdx.x;
  const int cid = selFinal[row];
  const int s = cid / Wn, w = cid % Wn;
  const int eRaw = s + w;
  const int e = (eRaw < Tn - 1) ? eRaw : Tn - 1;
  const int wi = e - s, L = wi + 1;
  __shared__ float wls[32];
  if (threadIdx.x < 32) {
    const int lane = threadIdx.x;
    float a = (lane < L) ? ta[s + lane] : -INFINITY;
    const float m = waveMax(a);
    const float ex = (lane < L) ? __expf(a - m) : 0.0f;
    const float ssum = waveSum(ex);
    wls[lane] = ex / ssum;                    // 0 for lane >= L
  }
  __syncthreads();
  float* dst = topEmb + (size_t)row * SPANn;
  for (int d = threadIdx.x; d < En; d += blockDim.x) {
    dst[d]      = te[(size_t)s * En + d];
    dst[En + d] = te[(size_t)e * En + d];
    float acc = 0.0f;
#pragma unroll
    for (int t = 0; t < Wn; ++t) {
      const int tt = (s + t < Tn - 1) ? (s + t) : (Tn - 1);
      acc += wls[t] * te[(size_t)tt * En + d];
    }
    dst[2 * En + Fn + d] = acc;
  }
  if (threadIdx.x < Fn) dst[2 * En + threadIdx.x] = esw[wi * Fn + threadIdx.x];
}

// ---------------- final masked combine ----------------
__global__ void final_k(const float* __restrict__ topScores,
                        const float* __restrict__ pair,
                        float* __restrict__ out) {
  const int idx = blockIdx.x * blockDim.x + threadIdx.x;
  if (idx >= KTOP * KTOP) return;
  const int i = idx / KTOP, j = idx % KTOP;
  out[idx] = (i > j)
      ? (topScores[i] + topScores[j] + pair[(size_t)i * PAIRLD + j])
      : -INFINITY;
}

// ---------------- host-side launcher ----------------
extern "C" void kernel_launch(void* const* d_in, const int* in_sizes, int n_in,
                              void* d_out, int out_size, void* d_ws, size_t ws_size,
                              hipStream_t stream) {
  const float* token_emb = (const float*)d_in[0];
  const float* attn_w    = (const float*)d_in[1];
  const float* attn_b    = (const float*)d_in[2];
  const float* esw       = (const float*)d_in[3];
  const float* eswp      = (const float*)d_in[4];
  const float* ms_w1     = (const float*)d_in[5];
  const float* ms_b1     = (const float*)d_in[6];
  const float* ms_w2     = (const float*)d_in[7];
  const float* ms_b2     = (const float*)d_in[8];
  const float* ws_w1     = (const float*)d_in[9];
  const float* ws_b1     = (const float*)d_in[10];
  const float* ws_w2     = (const float*)d_in[11];
  const float* ws_b2     = (const float*)d_in[12];
  const float* coarse_w  = (const float*)d_in[13];
  const float* coarse_b  = (const float*)d_in[14];
  const int*   smap      = (const int*)d_in[15];
  float* out = (float*)d_out;

  // workspace layout (floats)
  float* ws = (float*)d_ws;
  size_t off = 0;
  float* ta      = ws + off; off += Tn;                 // token attn logits
  float* Qm      = ws + off; off += (size_t)Tn * Hn;    // token_emb @ W1[start]
  float* Rm      = ws + off; off += (size_t)Tn * Hn;    // token_emb @ W1[end]
  float* Pm      = ws + off; off += (size_t)Tn * Hn;    // token_emb @ W1[head]
  float* widthW  = ws + off; off += (size_t)Wn * Hn;
  float* wscore  = ws + off; off += 32;
  float* scores  = ws + off; off += NCAND;
  int*   order   = (int*)(ws + off); off += NCAND;
  int*   selRaw  = (int*)(ws + off); off += 512;
  int*   cnt     = (int*)(ws + off); off += 32;
  int*   selF    = (int*)(ws + off); off += 512;
  float* topS    = ws + off; off += 512;
  float* topE    = ws + off; off += (size_t)KTOP * SPANn;
  float* tmp     = ws + off; off += (size_t)KTOP * SPANn;
  float* pair    = ws + off; off += (size_t)PAIRLD * PAIRLD;
  (void)ws_size; (void)n_in; (void)in_sizes; (void)out_size;

  // macro-tile grid helper: 32x32 of C per wave, 4 waves (128 thr) per block
  auto gemmBlocks = [](int M, int N) {
    const int tiles = ((M + 31) / 32) * ((N + 31) / 32);
    return (tiles + 3) / 4;
  };

  // 1) token attention logits: 1024 waves
  token_attn_k<<<Tn / 8, 256, 0, stream>>>(token_emb, attn_w, attn_b, ta);

  // 2) Q/R/P = token_emb @ ms_w1 row-slices  (WMMA f32)
  gemm_wmma_f32<<<gemmBlocks(Tn, Hn), 128, 0, stream>>>(
      token_emb, ms_w1, nullptr, Qm, Tn, Hn, En, En, Hn, Hn, 0);
  gemm_wmma_f32<<<gemmBlocks(Tn, Hn), 128, 0, stream>>>(
      token_emb, ms_w1 + (size_t)En * Hn, nullptr, Rm, Tn, Hn, En, En, Hn, Hn, 0);
  gemm_wmma_f32<<<gemmBlocks(Tn, Hn), 128, 0, stream>>>(
      token_emb, ms_w1 + (size_t)(2 * En + Fn) * Hn, nullptr, Pm,
      Tn, Hn, En, En, Hn, Hn, 0);

  // 3) width feature projection + width-prior scores
  width_proj_k<<<Wn, 256, 0, stream>>>(esw, ms_w1, widthW);
  wscore_k<<<Wn, 256, 0, stream>>>(eswp, ws_w1, ws_b1, ws_w2, ws_b2, wscore);

  // 4) fused per-span mention scores (wave per span)
  span_score_k<<<NCAND / 8, 256, 0, stream>>>(ta, Qm, Rm, Pm, widthW, wscore,
                                              ms_b1, ms_w2, ms_b2, smap, scores);

  // 5) full argsort in 256 KB LDS (CDNA5: 320 KB per WGP)
  sort_k<<<1, 1024, NSORT * sizeof(unsigned long long), stream>>>(scores, order);

  // 6) sequential greedy NMS (single wave)
  nms_k<<<1, 32, 0, stream>>>(order, smap, selRaw, cnt);

  // 7) sort winners by (start, end), pad shortfall
  selsort_k<<<1, 512, 0, stream>>>(selRaw, cnt, scores, selF, topS);

  // 8) rebuild the 409 winner embeddings
  topemb_k<<<KTOP, 256, 0, stream>>>(selF, token_emb, ta, esw, topE);

  // 9) tmp = top_emb @ coarse_w + coarse_b   [409 x 2324]
  gemm_wmma_f32<<<gemmBlocks(KTOP, SPANn), 128, 0, stream>>>(
      topE, coarse_w, coarse_b, tmp, KTOP, SPANn, SPANn, SPANn, SPANn, SPANn, 0);

  // 10) pair = tmp @ top_emb^T   [409 x 409]
  gemm_wmma_f32<<<gemmBlocks(KTOP, KTOP), 128, 0, stream>>>(
      tmp, topE, nullptr, pair, KTOP, KTOP, SPANn, SPANn, SPANn, PAIRLD, 1);

  // 11) masked antecedent combine
  final_k<<<(KTOP * KTOP + 255) / 256, 256, 0, stream>>>(topS, pair, out);
}